// BoundaryLoss_73246372266594
// MI455X (gfx1250) — compile-verified
//
#include <hip/hip_runtime.h>
#include <hip/hip_bf16.h>

// BoundaryLoss: exact EDT (two-pass separable) on pred/target masks, then
// mean |d_pred - d_targ|. B=8, H=W=256. Output: 1 float.

#define EDT_INF 1.0e6f
#define B_ 8
#define H_ 256
#define W_ 256

typedef __attribute__((ext_vector_type(2))) float v2f;
typedef __attribute__((ext_vector_type(8))) float v8f;

// Deterministic 256-thread block sum using V_WMMA_F32_16X16X4_F32.
// Each wave32: A(16x4) holds one diff per lane in VGPR0 (lanes 0-15 -> K=0,
// lanes 16-31 -> K=2), VGPR1 = 0. B(4x16) = all ones. D[m][n] = rowsum_m,
// accumulated in f32 (exact). Lane L<16 sums rows 0-7, lane L>=16 rows 8-15;
// wave total = local(lane0) + local(lane16). Wave sums combined in fixed
// order through LDS. Valid result returned on thread 0 only.
__device__ __forceinline__ float block_reduce_sum_256(float v, float* s_red) {
    v2f a; a.x = v;    a.y = 0.0f;
    v2f b; b.x = 1.0f; b.y = 1.0f;
    v8f c = {};
    v8f d = __builtin_amdgcn_wmma_f32_16x16x4_f32(
        /*neg_a=*/false, a, /*neg_b=*/false, b,
        /*c_mod=*/(short)0, c, /*reuse_a=*/false, /*reuse_b=*/false);
    float local = d[0] + d[1] + d[2] + d[3] + d[4] + d[5] + d[6] + d[7];
    float wave_sum = __shfl(local, 0, 32) + __shfl(local, 16, 32);
    int lane = (int)threadIdx.x & 31;
    int wave = (int)threadIdx.x >> 5;
    if (lane == 0) s_red[wave] = wave_sum;
    __syncthreads();
    float total = 0.0f;
    if (threadIdx.x == 0) {
        for (int w = 0; w < 8; ++w) total += s_red[w];   // fixed order
    }
    return total;
}

// Pass 1: per-column 1D distance (rows) to nearest background; writes g^2.
// grid = (B_, 2): blockIdx.y==0 -> pred mask, ==1 -> target mask.
// 256 threads = 256 columns; loads are row-coalesced across the wave.
__global__ __launch_bounds__(256)
void edt_vertical(const float* __restrict__ pred, const int* __restrict__ target,
                  float* __restrict__ g2p, float* __restrict__ g2t) {
    const int bimg  = blockIdx.x;
    const int which = blockIdx.y;
    const int j     = (int)threadIdx.x;
    const size_t img = (size_t)bimg * H_ * W_;
    float* g2 = (which ? g2t : g2p) + img;

    // down scan (top -> bottom)
    float prev = EDT_INF;
    for (int i = 0; i < H_; ++i) {
        const size_t idx = (size_t)i * W_ + j;
        bool bg = which ? (target[img + idx] == 0) : !(pred[img + idx] > 0.5f);
        float dcur = bg ? 0.0f : prev + 1.0f;
        prev = dcur;
        g2[idx] = dcur;                       // stash raw down-distance
    }
    // up scan (bottom -> top), combine, cap at INF, square
    prev = EDT_INF;
    for (int i = H_ - 1; i >= 0; --i) {
        const size_t idx = (size_t)i * W_ + j;
        bool bg = which ? (target[img + idx] == 0) : !(pred[img + idx] > 0.5f);
        float ucur = bg ? 0.0f : prev + 1.0f;
        prev = ucur;
        float g = fminf(fminf(g2[idx], ucur), EDT_INF);
        g2[idx] = g * g;
    }
}

// Pass 2 + diff: one block per (b, i) row. D2[j] = min_j' g2[j'] + (j-j')^2
// for both masks; |sqrt(p) - sqrt(t)|; WMMA block reduction -> row partial.
__global__ __launch_bounds__(256)
void edt_horizontal_diff(const float* __restrict__ g2p, const float* __restrict__ g2t,
                         float* __restrict__ row_sums) {
    __shared__ float s_gp[W_];
    __shared__ float s_gt[W_];
    __shared__ float s_red[8];

    const int row = (int)blockIdx.x;           // 0 .. B_*H_-1
    const int j   = (int)threadIdx.x;
    const size_t base = (size_t)row * W_;

    s_gp[j] = g2p[base + j];
    s_gt[j] = g2t[base + j];
    __syncthreads();

    const float jf = (float)j;
    float mp = 3.0e38f, mt = 3.0e38f;
#pragma unroll 8
    for (int jp = 0; jp < W_; ++jp) {
        float off = jf - (float)jp;
        float o2  = off * off;                 // LDS broadcast reads below: conflict-free
        mp = fminf(mp, o2 + s_gp[jp]);
        mt = fminf(mt, o2 + s_gt[jp]);
    }
    float diff  = fabsf(sqrtf(mp) - sqrtf(mt));
    float total = block_reduce_sum_256(diff, s_red);
    if (threadIdx.x == 0) row_sums[row] = total;
}

// Final: single block folds 2048 row partials (fixed order) and writes mean.
__global__ __launch_bounds__(256)
void final_reduce(const float* __restrict__ row_sums, int n,
                  float* __restrict__ out, float scale) {
    __shared__ float s_red[8];
    float v = 0.0f;
    for (int i = (int)threadIdx.x; i < n; i += 256) v += row_sums[i];
    float total = block_reduce_sum_256(v, s_red);
    if (threadIdx.x == 0) out[0] = total * scale;
}

extern "C" void kernel_launch(void* const* d_in, const int* in_sizes, int n_in,
                              void* d_out, int out_size, void* d_ws, size_t ws_size,
                              hipStream_t stream) {
    const float* pred  = (const float*)d_in[0];   // [8,1,256,256] f32
    const int*   target = (const int*)d_in[1];    // [8,1,256,256] i32

    const size_t plane = (size_t)B_ * H_ * W_;    // 524288
    float* g2p      = (float*)d_ws;               // g^2 for pred mask
    float* g2t      = g2p + plane;                // g^2 for target mask
    float* row_sums = g2t + plane;                // B_*H_ = 2048 partials

    dim3 g1(B_, 2);
    edt_vertical<<<g1, 256, 0, stream>>>(pred, target, g2p, g2t);
    edt_horizontal_diff<<<B_ * H_, 256, 0, stream>>>(g2p, g2t, row_sums);
    final_reduce<<<1, 256, 0, stream>>>(row_sums, B_ * H_, (float*)d_out,
                                        1.0f / (float)plane);
}